// GCN_70153995813500
// MI455X (gfx1250) — compile-verified
//
#include <hip/hip_runtime.h>

typedef float v2f __attribute__((ext_vector_type(2)));
typedef float v8f __attribute__((ext_vector_type(8)));

#define BN_EPS 1e-5f

// ---------------------------------------------------------------- degrees
__global__ __launch_bounds__(256) void init_deg(float* deg_out, float* deg_in, int n) {
    int i = blockIdx.x * blockDim.x + threadIdx.x;
    if (i < n) { deg_out[i] = 1.0f; deg_in[i] = 1.0f; }  // self-loop contribution
}

__global__ __launch_bounds__(256) void accum_deg(const int* __restrict__ src,
                                                 const int* __restrict__ dst,
                                                 float* deg_out, float* deg_in, int e) {
    int i = blockIdx.x * blockDim.x + threadIdx.x;
    if (i < e) {
        atomicAdd(&deg_out[src[i]], 1.0f);
        atomicAdd(&deg_in[dst[i]], 1.0f);
    }
}

__global__ __launch_bounds__(256) void finalize_norm(float* deg_out, float* deg_in, int n) {
    int i = blockIdx.x * blockDim.x + threadIdx.x;
    if (i < n) { deg_out[i] = rsqrtf(deg_out[i]); deg_in[i] = rsqrtf(deg_in[i]); }
}

// ------------------------------------------- hs = h * norm_src ; agg = hs (self-loop)
__global__ __launch_bounds__(256) void scale_init(const float* __restrict__ h,
                                                  const float* __restrict__ norm_src,
                                                  float* __restrict__ hs,
                                                  float* __restrict__ agg,
                                                  long total, int dlog) {
    long t = (long)blockIdx.x * blockDim.x + threadIdx.x;
    if (t >= total) return;
    long row = t >> dlog;
    float v = h[t] * norm_src[row];
    hs[t] = v;
    agg[t] = v;
}

// -------------------------------- agg[dst] += hs[src]  (L2-resident f32 atomics)
__global__ __launch_bounds__(256) void edge_aggregate(const int* __restrict__ src,
                                                      const int* __restrict__ dst,
                                                      const float* __restrict__ hs,
                                                      float* __restrict__ agg,
                                                      long total, int dlog) {
    long t = (long)blockIdx.x * blockDim.x + threadIdx.x;
    if (t >= total) return;
    const int cs = dlog - 2;                 // float4 chunks per row
    int e = (int)(t >> cs);
    int c = (int)(t & ((1 << cs) - 1));
    int si = src[e];
    int di = dst[e];
    const float4 v = *(const float4*)(hs + ((long)si << dlog) + (c << 2));
    float* o = agg + ((long)di << dlog) + (c << 2);
    atomicAdd(o + 0, v.x);
    atomicAdd(o + 1, v.y);
    atomicAdd(o + 2, v.z);
    atomicAdd(o + 3, v.w);
}

// ------------- out = act(rowscale * (A @ W) + bias), fp32 WMMA 16x16x4
// A: MxK row-major, W: KxNn row-major. Wave tile: 16 rows x 64 cols (4 accums).
// Block: 8 waves stacked -> 128 rows x 64 cols. M must be a multiple of 16.
// Out-of-range columns are load-clamped (memory-safe) and discarded at store,
// so the K-loop is branch-free and EXEC stays all-1s for every WMMA.
__global__ __launch_bounds__(256) void gemm_relu_wmma(const float* __restrict__ A,
                                                      const float* __restrict__ rowscale,
                                                      const float* __restrict__ W,
                                                      const float* __restrict__ bias,
                                                      float* __restrict__ out,
                                                      int M, int K, int Nn, int do_relu) {
    const int lane = threadIdx.x & 31;
    const int wave = threadIdx.x >> 5;
    const int mrow = lane & 15;
    const int kh   = lane >> 4;                    // 0 -> K{0,1}, 1 -> K{2,3}
    const int row0 = blockIdx.y * 128 + wave * 16;
    const int col0 = blockIdx.x * 64;
    if (row0 >= M) return;                         // wave-uniform exit: EXEC all-1s at WMMA

    const float* __restrict__ arow = A + (long)(row0 + mrow) * K + 2 * kh;

    // clamped (memory-safe) per-tile column indices, hoisted out of the loop
    const int lim = Nn - 1;
    const int c0 = min(col0 + 0 * 16 + mrow, lim);
    const int c1 = min(col0 + 1 * 16 + mrow, lim);
    const int c2 = min(col0 + 2 * 16 + mrow, lim);
    const int c3 = min(col0 + 3 * 16 + mrow, lim);

    const float* __restrict__ wk = W + (long)(2 * kh) * Nn;  // row (k + 2*kh) of W
    const long wstep = 4L * Nn;

    v8f acc0 = {}, acc1 = {}, acc2 = {}, acc3 = {};

#pragma unroll 2
    for (int k = 0; k < K; k += 4) {
        v2f a = *(const v2f*)(arow + k);           // A[row][k+2kh], A[row][k+2kh+1]
        v2f b0, b1, b2, b3;
        b0.x = wk[c0]; b0.y = wk[c0 + Nn];
        b1.x = wk[c1]; b1.y = wk[c1 + Nn];
        b2.x = wk[c2]; b2.y = wk[c2 + Nn];
        b3.x = wk[c3]; b3.y = wk[c3 + Nn];
        acc0 = __builtin_amdgcn_wmma_f32_16x16x4_f32(false, a, false, b0, (short)0, acc0, false, false);
        acc1 = __builtin_amdgcn_wmma_f32_16x16x4_f32(false, a, false, b1, (short)0, acc1, false, false);
        acc2 = __builtin_amdgcn_wmma_f32_16x16x4_f32(false, a, false, b2, (short)0, acc2, false, false);
        acc3 = __builtin_amdgcn_wmma_f32_16x16x4_f32(false, a, false, b3, (short)0, acc3, false, false);
        wk += wstep;
    }

    // per-output-row scale (rowscale ∘ (A@W) == (rowscale ∘ A)@W), applied once here
    float sr[8];
#pragma unroll
    for (int v = 0; v < 8; ++v) {
        int r = row0 + v + 8 * kh;                 // C/D layout: VGPR v -> M = v (+8 hi lanes)
        sr[v] = rowscale ? rowscale[r] : 1.0f;
    }

    v8f accs[4] = {acc0, acc1, acc2, acc3};
#pragma unroll
    for (int t = 0; t < 4; ++t) {
        int c = col0 + t * 16 + mrow;
        if (c >= Nn) continue;
        float bv = bias[c];
#pragma unroll
        for (int v = 0; v < 8; ++v) {
            int r = row0 + v + 8 * kh;
            float val = accs[t][v] * sr[v] + bv;
            if (do_relu) val = fmaxf(val, 0.0f);
            out[(long)r * Nn + c] = val;
        }
    }
}

// ---------------------------------------------------------- batchnorm stats
__global__ __launch_bounds__(256) void bn_stats(const float* __restrict__ z,
                                                float* __restrict__ mu,
                                                float* __restrict__ var,
                                                int M, int C) {
    __shared__ float ssum[256];
    __shared__ float ssq[256];
    const int c = blockIdx.x;                      // one block per column
    const int tid = threadIdx.x;
    float s = 0.0f, q = 0.0f;
    for (int r = tid; r < M; r += blockDim.x) {
        float v = z[(long)r * C + c];
        s += v; q += v * v;
    }
    ssum[tid] = s; ssq[tid] = q;
    __syncthreads();
    for (int off = 128; off > 0; off >>= 1) {
        if (tid < off) { ssum[tid] += ssum[tid + off]; ssq[tid] += ssq[tid + off]; }
        __syncthreads();
    }
    if (tid == 0) {
        float m = ssum[0] / (float)M;
        mu[c] = m;
        var[c] = ssq[0] / (float)M - m * m;        // population variance (jnp.var)
    }
}

// ----------------------------- out = BN(z) @ Wm2 + bm2  (CLS = 2)
__global__ __launch_bounds__(256) void bn_final(const float* __restrict__ z,
                                                const float* __restrict__ mu,
                                                const float* __restrict__ var,
                                                const float* __restrict__ gamma,
                                                const float* __restrict__ beta,
                                                const float* __restrict__ Wm2,
                                                const float* __restrict__ bm2,
                                                float* __restrict__ out, int M, int C) {
    __shared__ float sa[256];
    __shared__ float sb[256];
    for (int j = threadIdx.x; j < C; j += blockDim.x) {
        float a = gamma[j] * rsqrtf(var[j] + BN_EPS);
        sa[j] = a;
        sb[j] = beta[j] - mu[j] * a;
    }
    __syncthreads();
    int i = blockIdx.x * blockDim.x + threadIdx.x;
    if (i >= M) return;
    float acc0 = bm2[0], acc1 = bm2[1];
    const float* zr = z + (long)i * C;
    for (int j = 0; j < C; ++j) {
        float zh = zr[j] * sa[j] + sb[j];
        acc0 += zh * Wm2[j * 2 + 0];
        acc1 += zh * Wm2[j * 2 + 1];
    }
    out[i * 2 + 0] = acc0;
    out[i * 2 + 1] = acc1;
}

extern "C" void kernel_launch(void* const* d_in, const int* in_sizes, int n_in,
                              void* d_out, int out_size, void* d_ws, size_t ws_size,
                              hipStream_t stream) {
    const float* feat  = (const float*)d_in[0];
    const int*   src   = (const int*)d_in[1];
    const int*   dst   = (const int*)d_in[2];
    const float* Ws[3] = {(const float*)d_in[3], (const float*)d_in[5], (const float*)d_in[7]};
    const float* bs[3] = {(const float*)d_in[4], (const float*)d_in[6], (const float*)d_in[8]};
    const float* Wm1   = (const float*)d_in[9];
    const float* bm1   = (const float*)d_in[10];
    const float* gamma = (const float*)d_in[11];
    const float* beta  = (const float*)d_in[12];
    const float* Wm2   = (const float*)d_in[13];
    const float* bm2   = (const float*)d_in[14];

    const int N = in_sizes[0] / 128;
    const int E = in_sizes[1];
    const int HID = 200;

    // workspace layout (floats)
    float* ws      = (float*)d_ws;
    float* norm_s  = ws;                          // N  (deg_out -> rsqrt in place)
    float* norm_d  = norm_s + N;                  // N  (deg_in  -> rsqrt in place)
    float* hs      = norm_d + N;                  // N*256
    float* agg     = hs + (size_t)N * 256;        // N*256
    float* h       = agg + (size_t)N * 256;       // N*256
    float* mu      = h + (size_t)N * 256;         // 256
    float* var     = mu + 256;                    // 256
    float* z       = hs;                          // reuse hs for MLP hidden (N*200)

    const dim3 b256(256);

    // degree / symmetric norm
    init_deg<<<(N + 255) / 256, b256, 0, stream>>>(norm_s, norm_d, N);
    accum_deg<<<(E + 255) / 256, b256, 0, stream>>>(src, dst, norm_s, norm_d, E);
    finalize_norm<<<(N + 255) / 256, b256, 0, stream>>>(norm_s, norm_d, N);

    const int dins[3]  = {128, 128, 256};
    const int douts[3] = {128, 256, 256};
    const float* hin = feat;
    for (int l = 0; l < 3; ++l) {
        const int din  = dins[l];
        const int dout = douts[l];
        const int dlog = (din == 128) ? 7 : 8;

        long nt = (long)N << dlog;
        scale_init<<<(unsigned)((nt + 255) / 256), b256, 0, stream>>>(hin, norm_s, hs, agg, nt, dlog);

        long et = (long)E << (dlog - 2);
        edge_aggregate<<<(unsigned)((et + 255) / 256), b256, 0, stream>>>(src, dst, hs, agg, et, dlog);

        dim3 grid((dout + 63) / 64, (N + 127) / 128);
        gemm_relu_wmma<<<grid, b256, 0, stream>>>(agg, norm_d, Ws[l], bs[l], h, N, din, dout, 1);
        hin = h;
    }

    // MLP hidden: z = relu(h @ Wm1 + bm1)
    {
        dim3 grid((HID + 63) / 64, (N + 127) / 128);
        gemm_relu_wmma<<<grid, b256, 0, stream>>>(h, nullptr, Wm1, bm1, z, N, 256, HID, 1);
    }

    // batchnorm stats + fused BN/output linear
    bn_stats<<<HID, b256, 0, stream>>>(z, mu, var, N, HID);
    bn_final<<<(N + 255) / 256, b256, 0, stream>>>(z, mu, var, gamma, beta, Wm2, bm2,
                                                   (float*)d_out, N, HID);
}